// GCN_85332410237296
// MI455X (gfx1250) — compile-verified
//
#include <hip/hip_runtime.h>
#include <hip/hip_bf16.h>

typedef float v2f __attribute__((ext_vector_type(2)));
typedef float v8f __attribute__((ext_vector_type(8)));

#define WG 256

// ---------- helpers: order-preserving float <-> uint for atomic max ----------
__device__ __forceinline__ unsigned enc_f32(float f) {
  unsigned u = __float_as_uint(f);
  return (u & 0x80000000u) ? ~u : (u | 0x80000000u);
}
__device__ __forceinline__ float dec_f32(unsigned u) {
  u = (u & 0x80000000u) ? (u & 0x7FFFFFFFu) : ~u;
  return __uint_as_float(u);
}

// ---------- 0: workspace init (deg=1 self-loop, accumulators=0, gmax=0) -----
__global__ __launch_bounds__(WG) void k_init(float* deg, float* H1, float* Zagg,
                                             unsigned* genc, int N, int GC) {
  int i = blockIdx.x * WG + threadIdx.x;
  if (i < N * 16) H1[i] = 0.0f;
  if (i < N * 8) Zagg[i] = 0.0f;
  if (i < N) deg[i] = 1.0f;             // self loop
  if (i < GC) genc[i] = 0u;             // encodes "-inf" sentinel
}

// ---------- 1: in-degree accumulation --------------------------------------
__global__ __launch_bounds__(WG) void k_degree(const long long* __restrict__ dst,
                                               float* __restrict__ deg, int E) {
  int e = blockIdx.x * WG + threadIdx.x;
  if (e < E) atomicAdd(&deg[(int)dst[e]], 1.0f);
}

// ---------- 2: deg -> 1/sqrt(deg) in place ----------------------------------
__global__ __launch_bounds__(WG) void k_rsqrt(float* deg, int N) {
  int i = blockIdx.x * WG + threadIdx.x;
  if (i < N) deg[i] = rsqrtf(fmaxf(deg[i], 1.0f));
}

// ---------- 3: XW = X @ W1   [N,1433]x[1433,16], WMMA f32 16x16x4 -----------
// block = 256 threads = 8 waves; each wave computes a 16x16 output tile.
// Main K loop unguarded + software-pipelined; single guarded tail step.
#define KCHUNK 256
__global__ __launch_bounds__(WG) void k_gemm1(const float* __restrict__ x,
                                              const float* __restrict__ W1,
                                              float* __restrict__ XW,
                                              int N, int K) {
  __shared__ float ldsB[KCHUNK * 16];
  const int lane = threadIdx.x & 31;
  const int wave = threadIdx.x >> 5;
  const int m = lane & 15;        // row-in-tile (A) / col (B,C)
  const int hi = lane >> 4;       // half-select
  long long row0 = ((long long)blockIdx.x * 8 + wave) * 16;
  long long rowA = row0 + m;
  if (rowA > (long long)N - 1) rowA = (long long)N - 1;  // clamp; stores guarded
  // pre-offset by lane's K half: xrow[k] == x[rowA*K + 2*hi + k]
  const float* xrow = x + rowA * (long long)K + 2 * hi;

  const int K4 = K & ~3;
  const int K16 = K * 16;
  v8f acc = {0.f, 0.f, 0.f, 0.f, 0.f, 0.f, 0.f, 0.f};

  for (int kbase = 0; kbase < K4; kbase += KCHUNK) {
    __syncthreads();
    {   // vectorized staging: b128 global -> b128 LDS; scalar fallback at pad
      int gbase = kbase * 16;
      for (int i = threadIdx.x * 4; i < KCHUNK * 16; i += WG * 4) {
        int g = gbase + i;
        float4 v;
        if (g + 3 < K16) {
          v = *(const float4*)(W1 + g);
        } else {
          v.x = (g     < K16) ? W1[g]     : 0.0f;
          v.y = (g + 1 < K16) ? W1[g + 1] : 0.0f;
          v.z = (g + 2 < K16) ? W1[g + 2] : 0.0f;
          v.w = (g + 3 < K16) ? W1[g + 3] : 0.0f;
        }
        *(float4*)(ldsB + i) = v;
      }
    }
    __syncthreads();
    int kcnt = K4 - kbase;
    if (kcnt > KCHUNK) kcnt = KCHUNK;
    int kcnt8 = kcnt & ~7;
    const float* ap = xrow + kbase;
    const float* bp = ldsB + 2 * hi * 16 + m;  // bp[k*16] == lds[(k+2hi)*16+m]
    int k = 0;
    if (kcnt8 >= 8) {
      // software pipeline: prefetch next iteration's operands before WMMAs
      v2f a0, a1, b0, b1;
      a0.x = ap[0]; a0.y = ap[1];
      a1.x = ap[4]; a1.y = ap[5];
      b0.x = bp[0];  b0.y = bp[16];
      b1.x = bp[64]; b1.y = bp[80];
      for (k = 0; k + 8 < kcnt8; k += 8) {
        v2f na0, na1, nb0, nb1;
        na0.x = ap[k + 8];  na0.y = ap[k + 9];
        na1.x = ap[k + 12]; na1.y = ap[k + 13];
        nb0.x = bp[(k + 8) * 16];   nb0.y = bp[(k + 8) * 16 + 16];
        nb1.x = bp[(k + 12) * 16];  nb1.y = bp[(k + 12) * 16 + 16];
        acc = __builtin_amdgcn_wmma_f32_16x16x4_f32(
            false, a0, false, b0, (short)0, acc, false, false);
        acc = __builtin_amdgcn_wmma_f32_16x16x4_f32(
            false, a1, false, b1, (short)0, acc, false, false);
        a0 = na0; a1 = na1; b0 = nb0; b1 = nb1;
      }
      acc = __builtin_amdgcn_wmma_f32_16x16x4_f32(
          false, a0, false, b0, (short)0, acc, false, false);
      acc = __builtin_amdgcn_wmma_f32_16x16x4_f32(
          false, a1, false, b1, (short)0, acc, false, false);
      k += 8;
    }
    for (; k < kcnt; k += 4) {                 // at most one 4-step
      v2f a, b;
      a.x = ap[k]; a.y = ap[k + 1];
      b.x = bp[k * 16]; b.y = bp[k * 16 + 16];
      acc = __builtin_amdgcn_wmma_f32_16x16x4_f32(
          false, a, false, b, (short)0, acc, false, false);
    }
  }

  // K tail (K % 4 != 0): the ONLY guarded step; B read straight from global.
  if (K4 < K) {
    int kg = K4 + 2 * hi;
    v2f a, b;
    a.x = (kg < K)     ? xrow[K4]     : 0.0f;
    a.y = (kg + 1 < K) ? xrow[K4 + 1] : 0.0f;
    b.x = (kg < K)     ? W1[kg * 16 + m]       : 0.0f;
    b.y = (kg + 1 < K) ? W1[(kg + 1) * 16 + m] : 0.0f;
    acc = __builtin_amdgcn_wmma_f32_16x16x4_f32(
        false, a, false, b, (short)0, acc, false, false);
  }

  // C layout: VGPR v, half hi -> M = row0 + v + 8*hi, col = m
  float* outp = XW + (row0 + 8 * hi) * 16 + m;
  if (row0 + 16 <= (long long)N) {             // wave-uniform fast path
#pragma unroll
    for (int v = 0; v < 8; ++v) outp[v * 16] = acc[v];
  } else {
#pragma unroll
    for (int v = 0; v < 8; ++v)
      if (row0 + 8 * hi + v < (long long)N) outp[v * 16] = acc[v];
  }
}

// ---------- 4: edge aggregation layer 1 (16 lanes per edge) -----------------
__global__ __launch_bounds__(WG) void k_agg1(const long long* __restrict__ src,
                                             const long long* __restrict__ dst,
                                             const float* __restrict__ dinv,
                                             const float* __restrict__ XW,
                                             float* __restrict__ H1, int E) {
  unsigned idx = blockIdx.x * WG + threadIdx.x;
  unsigned e = idx >> 4;
  int t = (int)(idx & 15u);
  if (e < (unsigned)E) {
    int s = (int)src[e], d = (int)dst[e];
    float c = dinv[s] * dinv[d];
    atomicAdd(&H1[d * 16 + t], XW[s * 16 + t] * c);
  }
}

// ---------- 5: H1 = relu(H1 + XW*dinv^2 + b1) -------------------------------
__global__ __launch_bounds__(WG) void k_fin1(float* __restrict__ H1,
                                             const float* __restrict__ XW,
                                             const float* __restrict__ dinv,
                                             const float* __restrict__ b1, int N) {
  int idx = blockIdx.x * WG + threadIdx.x;
  if (idx < N * 16) {
    int n = idx >> 4, j = idx & 15;
    float dv = dinv[n];
    H1[idx] = fmaxf(H1[idx] + XW[idx] * dv * dv + b1[j], 0.0f);
  }
}

// ---------- 6: ZW = H1 @ W2   [N,16]x[16,7->16 padded], 4 WMMAs/tile --------
__global__ __launch_bounds__(WG) void k_gemm2(const float* __restrict__ H1,
                                              const float* __restrict__ W2,
                                              float* __restrict__ ZW,
                                              int N, int C) {
  __shared__ float ldsB[16 * 16];
  {
    int i = threadIdx.x;
    if (i < 256) {
      int k = i >> 4, n = i & 15;
      ldsB[i] = (n < C) ? W2[k * C + n] : 0.0f;
    }
  }
  __syncthreads();
  const int lane = threadIdx.x & 31;
  const int wave = threadIdx.x >> 5;
  const int m = lane & 15;
  const int hi = lane >> 4;
  long long row0 = ((long long)blockIdx.x * 8 + wave) * 16;
  long long rowA = row0 + m;
  if (rowA > (long long)N - 1) rowA = (long long)N - 1;
  const float* arow = H1 + rowA * 16 + 2 * hi;
  const float* bp = ldsB + 2 * hi * 16 + m;

  v8f acc = {0.f, 0.f, 0.f, 0.f, 0.f, 0.f, 0.f, 0.f};
#pragma unroll
  for (int k = 0; k < 16; k += 4) {
    v2f a, b;
    a.x = arow[k];      a.y = arow[k + 1];
    b.x = bp[k * 16];   b.y = bp[k * 16 + 16];
    acc = __builtin_amdgcn_wmma_f32_16x16x4_f32(
        false, a, false, b, (short)0, acc, false, false);
  }

  float* outp = ZW + (row0 + 8 * hi) * 16 + m;
  if (row0 + 16 <= (long long)N) {             // wave-uniform fast path
#pragma unroll
    for (int v = 0; v < 8; ++v) outp[v * 16] = acc[v];
  } else {
#pragma unroll
    for (int v = 0; v < 8; ++v)
      if (row0 + 8 * hi + v < (long long)N) outp[v * 16] = acc[v];
  }
}

// ---------- 7: edge aggregation layer 2 (8 lanes per edge, col7 is zero) ----
__global__ __launch_bounds__(WG) void k_agg2(const long long* __restrict__ src,
                                             const long long* __restrict__ dst,
                                             const float* __restrict__ dinv,
                                             const float* __restrict__ ZW,
                                             float* __restrict__ Zagg, int E) {
  unsigned idx = blockIdx.x * WG + threadIdx.x;
  unsigned e = idx >> 3;
  int t = (int)(idx & 7u);
  if (e < (unsigned)E) {
    int s = (int)src[e], d = (int)dst[e];
    float c = dinv[s] * dinv[d];
    atomicAdd(&Zagg[d * 8 + t], ZW[s * 16 + t] * c);
  }
}

// ---------- 8: logits, log_softmax -> out, atomicMax pooling ----------------
__global__ __launch_bounds__(WG) void k_fin2(const float* __restrict__ Zagg,
                                             const float* __restrict__ ZW,
                                             const float* __restrict__ dinv,
                                             const float* __restrict__ b2,
                                             const long long* __restrict__ batch,
                                             unsigned* __restrict__ genc,
                                             float* __restrict__ out_ls,
                                             int N, int C) {
  int n = blockIdx.x * WG + threadIdx.x;
  if (n >= N) return;
  float dv = dinv[n];
  float dv2 = dv * dv;
  float z[7];
  float mx = -3.402823466e+38f;
#pragma unroll
  for (int c = 0; c < 7; ++c) {
    float v = Zagg[n * 8 + c] + ZW[n * 16 + c] * dv2 + b2[c];
    z[c] = v;
    mx = fmaxf(mx, v);
  }
  float ssum = 0.0f;
#pragma unroll
  for (int c = 0; c < 7; ++c) ssum += expf(z[c] - mx);
  float lse = mx + logf(ssum);
  int b = (int)batch[n];
#pragma unroll
  for (int c = 0; c < 7; ++c) {
    out_ls[n * 7 + c] = z[c] - lse;
    atomicMax(&genc[b * 7 + c], enc_f32(z[c]));
  }
}

// ---------- 9: decode pooled max into first 448 outputs ---------------------
__global__ __launch_bounds__(WG) void k_writeg(const unsigned* __restrict__ genc,
                                               float* __restrict__ out, int GC) {
  int i = blockIdx.x * WG + threadIdx.x;
  if (i < GC) out[i] = dec_f32(genc[i]);
}

// ============================================================================
extern "C" void kernel_launch(void* const* d_in, const int* in_sizes, int n_in,
                              void* d_out, int out_size, void* d_ws, size_t ws_size,
                              hipStream_t stream) {
  const float*     x     = (const float*)d_in[0];
  const long long* ei    = (const long long*)d_in[1];
  const long long* batch = (const long long*)d_in[2];
  const float*     W1    = (const float*)d_in[3];
  const float*     b1    = (const float*)d_in[4];
  const float*     W2    = (const float*)d_in[5];
  const float*     b2    = (const float*)d_in[6];

  const int N = in_sizes[2];               // 100000
  const int E = in_sizes[1] / 2;           // 3200000
  const int K = in_sizes[0] / N;           // 1433
  const int C = in_sizes[6];               // 7
  const int GC = out_size - N * C;         // 64*7 = 448

  const long long* src = ei;
  const long long* dst = ei + E;

  // ---- workspace carve-up (256B aligned) ----
  auto align_up = [](size_t v) { return (v + 255) & ~(size_t)255; };
  char* p = (char*)d_ws;
  float* deg  = (float*)p;  p += align_up((size_t)N * 4);           // reused as dinv
  float* XW   = (float*)p;  p += align_up((size_t)N * 16 * 4);
  float* H1   = (float*)p;  p += align_up((size_t)N * 16 * 4);
  float* ZW   = (float*)p;  p += align_up((size_t)N * 16 * 4);
  float* Zagg = (float*)p;  p += align_up((size_t)N * 8 * 4);
  unsigned* genc = (unsigned*)p;

  float* out   = (float*)d_out;   // [0,448): pooled max
  float* outls = out + GC;        // [448,...): log_softmax, [N,7]

  const long long NT = (long long)N * 16;
  dim3 blk(WG);

  k_init<<<dim3((unsigned)((NT + WG - 1) / WG)), blk, 0, stream>>>(
      deg, H1, Zagg, genc, N, GC);
  k_degree<<<dim3((unsigned)((E + WG - 1) / WG)), blk, 0, stream>>>(dst, deg, E);
  k_rsqrt<<<dim3((unsigned)((N + WG - 1) / WG)), blk, 0, stream>>>(deg, N);

  // 8 waves/block, 16 rows/wave -> 128 rows/block
  unsigned gtiles = (unsigned)((N + 127) / 128);
  k_gemm1<<<dim3(gtiles), blk, 0, stream>>>(x, W1, XW, N, K);

  k_agg1<<<dim3((unsigned)(((long long)E * 16 + WG - 1) / WG)), blk, 0, stream>>>(
      src, dst, deg, XW, H1, E);
  k_fin1<<<dim3((unsigned)((NT + WG - 1) / WG)), blk, 0, stream>>>(
      H1, XW, deg, b1, N);

  k_gemm2<<<dim3(gtiles), blk, 0, stream>>>(H1, W2, ZW, N, C);

  k_agg2<<<dim3((unsigned)(((long long)E * 8 + WG - 1) / WG)), blk, 0, stream>>>(
      src, dst, deg, ZW, Zagg, E);
  k_fin2<<<dim3((unsigned)((N + WG - 1) / WG)), blk, 0, stream>>>(
      Zagg, ZW, deg, b2, batch, genc, outls, N, C);
  k_writeg<<<dim3((unsigned)((GC + WG - 1) / WG)), blk, 0, stream>>>(genc, out, GC);
}